// VAELoss_88064009437959
// MI455X (gfx1250) — compile-verified
//
#include <hip/hip_runtime.h>
#include <hip/hip_bf16.h>

typedef __attribute__((ext_vector_type(2))) float v2f;
typedef __attribute__((ext_vector_type(8))) float v8f;

#define B_   8
#define A_   32
#define S_   16
#define D3_  58
#define TDE  18750        // T*DE  (emotion row length)
#define TD3  43500        // T*D3  (3dmm row length)
#define KTOT 62250        // virtual K: [0,43500) = 3dmm, [43500,62250) = emotion
#define MEND 43500
#define CH   128          // K-chunk staged in LDS
#define NCH  487          // ceil(KTOT/CH)
#define KSPLIT 32         // K-slices per batch element
#define CPB  16           // chunks per block = ceil(NCH/KSPLIT)
#define PSTR 132          // padded LDS row stride (132 mod 64 = 4 -> conflict-free)

#define WE (1.0f / 18750.0f)   // emotion mean weight
#define W1 (1.0f / 39000.0f)   // 3dmm dims 0..51
#define W2 (10.0f / 4500.0f)   // 3dmm dims 52..57 (x10)

// workspace layout (bytes): [0,16) sums, [64,576) idx, [1024,2048) gnorm, [2048,18432) cross
#define WS_SUMS   0
#define WS_IDX    64
#define WS_GNORM  1024
#define WS_CROSS  2048
#define WS_ZFLOAT 4608          // floats to zero (18432 B)

// ---------------------------------------------------------------------------
// Kernel 1: partial cost GEMM via WMMA f32 16x16x4.
// Grid = B*KSPLIT blocks of 256 threads (8 wave32). P tile staged with
// async global->LDS copies (ASYNCcnt); G tile staged through registers so
// the mean-weights can be folded in and gnorm accumulated on the fly.
// ---------------------------------------------------------------------------
__launch_bounds__(256)
__global__ void cost_partial(const float* __restrict__ ge, const float* __restrict__ g3,
                             const float* __restrict__ pe, const float* __restrict__ p3,
                             float* __restrict__ gnormW, float* __restrict__ crossW)
{
    __shared__ float Pch[16 * PSTR];
    __shared__ float Gch[32 * PSTR];
    __shared__ float redbuf[8 * 512];
    __shared__ float gpart[256];

    const int b     = blockIdx.x / KSPLIT;
    const int slice = blockIdx.x % KSPLIT;
    const int c0    = slice * CPB;
    const int c1    = min(NCH, c0 + CPB);

    const int tid  = threadIdx.x;
    const int lane = tid & 31;
    const int wave = tid >> 5;

    const int ps  = tid >> 4;             // pred row 0..15, 8 floats/chunk
    const int pcb = (tid & 15) * 8;
    const int ga  = tid >> 3;             // gt row 0..31, 16 floats/chunk
    const int gcb = (tid & 7) * 16;

    const float* perow = pe + (b * S_ + ps) * TDE;
    const float* p3row = p3 + (b * S_ + ps) * TD3;
    const float* gerow = ge + (b * A_ + ga) * TDE;
    const float* g3row = g3 + (b * A_ + ga) * TD3;

    // LDS byte address of this thread's P-tile slot: generic pointers to
    // __shared__ carry the LDS offset in their low 32 bits (flat aperture rule).
    const unsigned ldsP = (unsigned)(unsigned long long)(uintptr_t)&Pch[ps * PSTR + pcb];

    const int m  = lane & 15;             // WMMA frag row (A) / column (B)
    const int hi = (lane >> 4) << 1;      // lanes 16-31 hold K+2,K+3

    v8f acc0 = {0.f,0.f,0.f,0.f,0.f,0.f,0.f,0.f};   // tile a = 0..15
    v8f acc1 = {0.f,0.f,0.f,0.f,0.f,0.f,0.f,0.f};   // tile a = 16..31
    float gnacc = 0.f;

    for (int c = c0; c < c1; ++c) {
        const int k0 = c * CH;
        if (k0 + CH <= MEND) {
            // ---- pure 3dmm chunk
            // P: async global->LDS, 32B per thread (16B-aligned: stride 174000B)
            unsigned long long gaP = (unsigned long long)(uintptr_t)(p3row + k0 + pcb);
            asm volatile("global_load_async_to_lds_b128 %0, %1, off\n\t"
                         "global_load_async_to_lds_b128 %0, %1, off offset:16"
                         :: "v"(ldsP), "v"(gaP) : "memory");

            // G: register path with weight folding + gnorm accumulation
            const float4* g4 = (const float4*)(g3row + k0 + gcb);
            __builtin_prefetch(g3row + k0 + gcb + CH, 0, 1);
            int d0 = (k0 + gcb) % D3_;
            #pragma unroll
            for (int i = 0; i < 4; ++i) {
                float4 v = g4[i];
                float in[4] = {v.x, v.y, v.z, v.w};
                float out[4];
                #pragma unroll
                for (int j = 0; j < 4; ++j) {
                    int d = d0 + i * 4 + j; if (d >= D3_) d -= D3_;
                    float w = (d < 52) ? W1 : W2;
                    float x = in[j];
                    gnacc += w * x * x;
                    out[j] = w * x;
                }
                float4 o; o.x = out[0]; o.y = out[1]; o.z = out[2]; o.w = out[3];
                *(float4*)&Gch[ga * PSTR + gcb + i * 4] = o;
            }
        } else if (k0 >= MEND && k0 + CH <= KTOT) {
            // ---- pure emotion chunk (rows only 8B-aligned: stride 75000B)
            const int q0 = k0 - MEND;
            unsigned long long gaP = (unsigned long long)(uintptr_t)(perow + q0 + pcb);
            asm volatile("global_load_async_to_lds_b64 %0, %1, off\n\t"
                         "global_load_async_to_lds_b64 %0, %1, off offset:8\n\t"
                         "global_load_async_to_lds_b64 %0, %1, off offset:16\n\t"
                         "global_load_async_to_lds_b64 %0, %1, off offset:24"
                         :: "v"(ldsP), "v"(gaP) : "memory");

            const float2* g2 = (const float2*)(gerow + q0 + gcb);
            __builtin_prefetch(gerow + q0 + gcb + CH, 0, 1);
            #pragma unroll
            for (int i = 0; i < 8; ++i) {
                float2 v = g2[i];
                gnacc += WE * (v.x * v.x + v.y * v.y);
                float2 o; o.x = WE * v.x; o.y = WE * v.y;
                *(float2*)&Gch[ga * PSTR + gcb + 2 * i] = o;
            }
        } else {
            // ---- boundary / tail chunk (at most twice per grid): guarded path
            #pragma unroll
            for (int i = 0; i < 8; ++i) {
                int k = k0 + pcb + i;
                float v = 0.f;
                if (k < MEND)      v = p3row[k];
                else if (k < KTOT) v = perow[k - MEND];
                Pch[ps * PSTR + pcb + i] = v;
            }
            #pragma unroll
            for (int i = 0; i < 16; ++i) {
                int k = k0 + gcb + i;
                float v = 0.f, w = 0.f;
                if (k < MEND)      { int d = k % D3_; w = (d < 52) ? W1 : W2; v = g3row[k]; }
                else if (k < KTOT) { w = WE; v = gerow[k - MEND]; }
                gnacc += w * v * v;
                Gch[ga * PSTR + gcb + i] = w * v;
            }
        }
        // drain this wave's async copies, then make the tile block-visible
        asm volatile("s_wait_asynccnt 0" ::: "memory");
        __syncthreads();

        const int kb = wave * 16;         // each wave owns 16 K of the chunk
        #pragma unroll
        for (int kk = 0; kk < 16; kk += 4) {
            const float* pa  = &Pch[m * PSTR + kb + kk + hi];
            const float* gb0 = &Gch[m * PSTR + kb + kk + hi];
            const float* gb1 = &Gch[(m + 16) * PSTR + kb + kk + hi];
            v2f af;  af.x  = pa[0];  af.y  = pa[1];
            v2f bf0; bf0.x = gb0[0]; bf0.y = gb0[1];
            v2f bf1; bf1.x = gb1[0]; bf1.y = gb1[1];
            acc0 = __builtin_amdgcn_wmma_f32_16x16x4_f32(false, af, false, bf0,
                                                         (short)0, acc0, false, false);
            acc1 = __builtin_amdgcn_wmma_f32_16x16x4_f32(false, af, false, bf1,
                                                         (short)0, acc1, false, false);
        }
        __syncthreads();
    }

    // dump per-wave partial tiles + per-thread gnorm partials
    #pragma unroll
    for (int i = 0; i < 8; ++i) {
        redbuf[wave * 512 + i * 32 + lane]       = acc0[i];
        redbuf[wave * 512 + 256 + i * 32 + lane] = acc1[i];
    }
    gpart[tid] = gnacc;
    __syncthreads();

    if (tid < 32) {
        float s = 0.f;
        #pragma unroll
        for (int j = 0; j < 8; ++j) s += gpart[tid * 8 + j];
        atomicAdd(&gnormW[b * A_ + tid], s);
    }

    // reduce 8 waves -> cross[s][a] partial, atomic into ws
    for (int e = tid; e < 512; e += 256) {
        float s = 0.f;
        #pragma unroll
        for (int w = 0; w < 8; ++w) s += redbuf[w * 512 + e];
        int tile = e >> 8;
        int i    = (e >> 5) & 7;
        int ln   = e & 31;
        int srow = i + ((ln >> 4) << 3);          // D-layout: M = vgpr + 8*(lane>=16)
        int acol = (ln & 15) + (tile << 4);
        atomicAdd(&crossW[b * 512 + srow * 32 + acol], s);
    }
}

// ---------------------------------------------------------------------------
// Kernel 2: argmin over a, per (b,s).
// ---------------------------------------------------------------------------
__global__ void argmin_kernel(const float* __restrict__ gnormW,
                              const float* __restrict__ crossW, int* __restrict__ idx)
{
    const int t = threadIdx.x;            // 0..127
    const int b = t >> 4;
    const float* cr = crossW + b * 512 + (t & 15) * 32;
    const float* gn = gnormW + b * A_;
    float best = gn[0] - 2.0f * cr[0];
    int bi = 0;
    for (int a = 1; a < A_; ++a) {
        float c = gn[a] - 2.0f * cr[a];
        if (c < best) { best = c; bi = a; }
    }
    idx[t] = bi;
}

// ---------------------------------------------------------------------------
// Kernel 3: gathered reconstruction sums (one block per pred row), float2 loads.
// ---------------------------------------------------------------------------
__launch_bounds__(256)
__global__ void rec_sums(const float* __restrict__ ge, const float* __restrict__ g3,
                         const float* __restrict__ pe, const float* __restrict__ p3,
                         const int* __restrict__ idx, float* __restrict__ sums)
{
    const int row = blockIdx.x;
    const int tid = threadIdx.x;
    const int grow = (row >> 4) * A_ + idx[row];
    const float* per = pe + row * TDE;
    const float* p3r = p3 + row * TD3;
    const float* ger = ge + grow * TDE;
    const float* g3r = g3 + grow * TD3;

    float se = 0.f, s1 = 0.f, s2 = 0.f;
    for (int k = tid * 2; k < TDE; k += 512) {
        float2 a = *(const float2*)(per + k);
        float2 g = *(const float2*)(ger + k);
        float dx = a.x - g.x, dy = a.y - g.y;
        se += dx * dx + dy * dy;
    }
    for (int k = tid * 2; k < TD3; k += 512) {
        float2 a = *(const float2*)(p3r + k);
        float2 g = *(const float2*)(g3r + k);
        float dx = a.x - g.x, dy = a.y - g.y;
        int d = k % D3_;
        if (d < 52) s1 += dx * dx; else s2 += dx * dx;
        int d2 = d + 1; if (d2 == D3_) d2 = 0;
        if (d2 < 52) s1 += dy * dy; else s2 += dy * dy;
    }

    __shared__ float r0[256], r1[256], r2[256];
    r0[tid] = se; r1[tid] = s1; r2[tid] = s2;
    __syncthreads();
    for (int off = 128; off > 0; off >>= 1) {
        if (tid < off) { r0[tid] += r0[tid+off]; r1[tid] += r1[tid+off]; r2[tid] += r2[tid+off]; }
        __syncthreads();
    }
    if (tid == 0) {
        atomicAdd(&sums[0], r0[0]);
        atomicAdd(&sums[1], r1[0]);
        atomicAdd(&sums[2], r2[0]);
    }
}

// ---------------------------------------------------------------------------
// Kernel 4: KL sum over 262144 elements, float2 loads.
// ---------------------------------------------------------------------------
__launch_bounds__(256)
__global__ void kld_sums(const float* __restrict__ mu, const float* __restrict__ sc,
                         float* __restrict__ sums)
{
    const int tid = threadIdx.x;
    const int N = 4 * 128 * 512;
    float acc = 0.f;
    for (int i = (blockIdx.x * 256 + tid) * 2; i < N; i += 128 * 256 * 2) {
        float2 m = *(const float2*)(mu + i);
        float2 s = *(const float2*)(sc + i);
        acc += 0.5f * (s.x * s.x + m.x * m.x - 1.0f) - __logf(s.x);
        acc += 0.5f * (s.y * s.y + m.y * m.y - 1.0f) - __logf(s.y);
    }
    __shared__ float r[256];
    r[tid] = acc;
    __syncthreads();
    for (int off = 128; off > 0; off >>= 1) {
        if (tid < off) r[tid] += r[tid + off];
        __syncthreads();
    }
    if (tid == 0) atomicAdd(&sums[3], r[0]);
}

__global__ void init_ws(float* ws) {
    int i = blockIdx.x * 256 + threadIdx.x;
    if (i < WS_ZFLOAT) ws[i] = 0.f;
}

__global__ void finalize(const float* __restrict__ sums, float* __restrict__ out) {
    float rec = sums[0] * (1.0f / (128.0f * 18750.0f))
              + sums[1] * (1.0f / (128.0f * 39000.0f))
              + sums[2] * (10.0f / (128.0f * 4500.0f));
    float kld = sums[3] * (1.0f / 262144.0f);
    out[0] = rec + 0.0002f * kld;
    out[1] = rec;
    out[2] = kld;
}

extern "C" void kernel_launch(void* const* d_in, const int* in_sizes, int n_in,
                              void* d_out, int out_size, void* d_ws, size_t ws_size,
                              hipStream_t stream) {
    (void)in_sizes; (void)n_in; (void)out_size; (void)ws_size;
    const float* ge = (const float*)d_in[0];
    const float* g3 = (const float*)d_in[1];
    const float* pe = (const float*)d_in[2];
    const float* p3 = (const float*)d_in[3];
    const float* mu = (const float*)d_in[4];
    const float* sc = (const float*)d_in[5];
    char* wsb = (char*)d_ws;
    float* sums   = (float*)(wsb + WS_SUMS);
    int*   idx    = (int*)(wsb + WS_IDX);
    float* gnormW = (float*)(wsb + WS_GNORM);
    float* crossW = (float*)(wsb + WS_CROSS);
    float* out    = (float*)d_out;

    hipLaunchKernelGGL(init_ws,       dim3(18),           dim3(256), 0, stream, (float*)wsb);
    hipLaunchKernelGGL(cost_partial,  dim3(B_ * KSPLIT),  dim3(256), 0, stream,
                       ge, g3, pe, p3, gnormW, crossW);
    hipLaunchKernelGGL(argmin_kernel, dim3(1),            dim3(128), 0, stream,
                       gnormW, crossW, idx);
    hipLaunchKernelGGL(rec_sums,      dim3(B_ * S_),      dim3(256), 0, stream,
                       ge, g3, pe, p3, idx, sums);
    hipLaunchKernelGGL(kld_sums,      dim3(128),          dim3(256), 0, stream, mu, sc, sums);
    hipLaunchKernelGGL(finalize,      dim3(1),            dim3(1),   0, stream, sums, out);
}